// QuantumTransformerForecaster_4844723110014
// MI455X (gfx1250) — compile-verified
//
#include <hip/hip_runtime.h>
#include <hip/hip_bf16.h>
#include <math.h>

#define DIM 4096
#define NQ  12
#define PI_F 3.14159265358979323846f

typedef float v2f __attribute__((ext_vector_type(2)));
typedef float v8f __attribute__((ext_vector_type(8)));

// ---------------------------------------------------------------------------
// Complex 2x2 gate on qubit q, state in LDS (re/im split arrays).
// Pairs (i0, i0|1<<q) are disjoint across threads -> in-place, barrier after.
// ---------------------------------------------------------------------------
__device__ __forceinline__ void apply_gate(float* re, float* im, int q,
    float g00r, float g00i, float g01r, float g01i,
    float g10r, float g10i, float g11r, float g11i, int tid)
{
    const int lo = 1 << q;
    for (int p = tid; p < (DIM / 2); p += 256) {
        int i0 = ((p >> q) << (q + 1)) | (p & (lo - 1));
        int i1 = i0 | lo;
        float a0r = re[i0], a0i = im[i0];
        float a1r = re[i1], a1i = im[i1];
        re[i0] = g00r * a0r - g00i * a0i + g01r * a1r - g01i * a1i;
        im[i0] = g00r * a0i + g00i * a0r + g01r * a1i + g01i * a1r;
        re[i1] = g10r * a0r - g10i * a0i + g11r * a1r - g11i * a1i;
        im[i1] = g10r * a0i + g10i * a0r + g11r * a1i + g11i * a1r;
    }
    __syncthreads();
}

// CX(control c, target t): gather new[i]=old[i^(1<<t)] when bit c set
// == swap i <-> i^(1<<t) over indices with bit_c=1, bit_t=0 (1024 disjoint swaps)
__device__ __forceinline__ void apply_cx(float* re, float* im, int c, int t, int tid)
{
    const int lo = (c < t) ? c : t;
    const int hi = (c < t) ? t : c;
    for (int p = tid; p < (DIM / 4); p += 256) {
        int a = ((p >> lo) << (lo + 1)) | (p & ((1 << lo) - 1));
        int i = ((a >> hi) << (hi + 1)) | (a & ((1 << hi) - 1));
        i |= (1 << c);            // control bit = 1, target bit = 0
        int j = i | (1 << t);
        float tr = re[i]; re[i] = re[j]; re[j] = tr;
        float ti = im[i]; im[i] = im[j]; im[j] = ti;
    }
    __syncthreads();
}

// ---------------------------------------------------------------------------
// Kernel 1: encode + reservoir. One (b,t) state per block, state in LDS.
// ---------------------------------------------------------------------------
__global__ __launch_bounds__(256) void reservoir_kernel(
    const float* __restrict__ seq, const float* __restrict__ resp,
    float* __restrict__ states)
{
    __shared__ float sre[DIM];
    __shared__ float sim[DIM];
    const int tid = threadIdx.x;
    const int bid = blockIdx.x;          // b*64 + t
    const float x = seq[bid];
    if (tid == 0) __builtin_prefetch(resp, 0, 3);

    for (int d = tid; d < DIM; d += 256) { sre[d] = 0.015625f; sim[d] = 0.0f; }
    __syncthreads();

    // encode: fused G = RZ(phi) * RY(theta) per qubit
    for (int i = 0; i < NQ; ++i) {
        float th = x * PI_F * (float)(i + 1) / 12.0f;
        float ph = x * PI_F / (float)(i + 1);
        float c = cosf(0.5f * th), s = sinf(0.5f * th);
        float cz = cosf(0.5f * ph), sz = sinf(0.5f * ph);
        apply_gate(sre, sim, i,
                   c * cz, -c * sz,   -s * cz,  s * sz,
                   s * cz,  s * sz,    c * cz,  c * sz, tid);
    }
    for (int i = 0; i < NQ - 1; ++i) apply_cx(sre, sim, i, i + 1, tid); // ladder
    apply_cx(sre, sim, NQ - 1, 0, tid);                                 // ring

    // reservoir: 3 layers of U3 + CZ + CX_LONG
    for (int layer = 0; layer < 3; ++layer) {
        for (int i = 0; i < NQ; ++i) {
            const float* pp = resp + (layer * NQ + i) * 3;
            float t = pp[0], p = pp[1], l = pp[2];
            float ct = cosf(0.5f * t), st = sinf(0.5f * t);
            float cp = cosf(p), sp = sinf(p);
            float cl = cosf(l), sl = sinf(l);
            float cpl = cosf(p + l), spl = sinf(p + l);
            apply_gate(sre, sim, i,
                       ct, 0.0f,          -cl * st, -sl * st,
                       cp * st, sp * st,   cpl * ct, spl * ct, tid);
        }
        int mask = (layer & 1) ? 0x2AA : 0x555;   // CZ pair positions
        for (int d = tid; d < DIM; d += 256) {
            int pr = d & (d >> 1) & mask;
            if (__popc(pr) & 1) { sre[d] = -sre[d]; sim[d] = -sim[d]; }
        }
        __syncthreads();
        apply_cx(sre, sim, 0, NQ - 1, tid);       // CX_LONG
    }

    float* out = states + (size_t)bid * (2 * DIM);
    for (int d = tid; d < DIM; d += 256) { out[d] = sre[d]; out[DIM + d] = sim[d]; }
}

// ---------------------------------------------------------------------------
// Kernel 2: projection branches + Z expectations.
// bid < 2048: (h, K/V) x 512 states. bid >= 2048: Q branches, t==63 only.
// ---------------------------------------------------------------------------
__global__ __launch_bounds__(256) void project_kernel(
    const float* __restrict__ states, const float* __restrict__ hp,
    float* __restrict__ qbuf, float* __restrict__ kbuf, float* __restrict__ vbuf)
{
    __shared__ float sre[DIM];
    __shared__ float sim[DIM];
    __shared__ float red[8][NQ];
    const int tid = threadIdx.x;
    const int bid = blockIdx.x;

    int h, which, sid;
    if (bid < 2048) {
        int br = bid >> 9;  sid = bid & 511;
        h = br >> 1;        which = (br & 1) ? 2 : 1;    // K=1, V=2
    } else {
        int i = bid - 2048; h = i >> 3;
        int b = i & 7;      sid = b * 64 + 63;  which = 0; // Q at t=63
    }

    const float* st = states + (size_t)sid * (2 * DIM);
    for (int d = tid; d < DIM; d += 256) { sre[d] = st[d]; sim[d] = st[DIM + d]; }
    __syncthreads();

    const float* pp = hp + (h * 3 + which) * 48;
    for (int layer = 0; layer < 2; ++layer) {
        for (int i = 0; i < NQ; ++i) {
            int k = (layer * NQ + i) * 2;
            float a = pp[k], b = pp[k + 1];
            float c = cosf(0.5f * a), s = sinf(0.5f * a);
            float cz = cosf(0.5f * b), sz = sinf(0.5f * b);
            apply_gate(sre, sim, i,
                       c * cz, -c * sz,  -s * cz,  s * sz,
                       s * cz,  s * sz,   c * cz,  c * sz, tid);
        }
        for (int i = 0; i < NQ - 1; ++i) apply_cx(sre, sim, i, i + 1, tid);
    }

    // Z expectations: out[q] = sum_d |amp|^2 * (1 - 2*bit_{11-q}(d))
    float acc[NQ];
    #pragma unroll
    for (int q = 0; q < NQ; ++q) acc[q] = 0.0f;
    for (int d = tid; d < DIM; d += 256) {
        float pr = sre[d] * sre[d] + sim[d] * sim[d];
        #pragma unroll
        for (int q = 0; q < NQ; ++q)
            acc[q] += ((d >> (11 - q)) & 1) ? -pr : pr;
    }
    #pragma unroll
    for (int q = 0; q < NQ; ++q)
        for (int off = 16; off > 0; off >>= 1)
            acc[q] += __shfl_xor(acc[q], off, 32);
    const int wv = tid >> 5, ln = tid & 31;
    if (ln == 0) {
        #pragma unroll
        for (int q = 0; q < NQ; ++q) red[wv][q] = acc[q];
    }
    __syncthreads();
    if (tid < NQ) {
        float s = 0.0f;
        for (int w = 0; w < 8; ++w) s += red[w][tid];   // fixed order: deterministic
        const int b = sid >> 6, t = sid & 63;
        if (which == 0)      qbuf[(h * 8 + b) * NQ + tid] = s;
        else if (which == 1) kbuf[((h * 8 + b) * 64 + t) * NQ + tid] = s;
        else                 vbuf[((h * 8 + b) * 64 + t) * NQ + tid] = s;
    }
}

// ---------------------------------------------------------------------------
// Kernel 3: last-row attention (VALU) + MLP via V_WMMA_F32_16X16X4_F32.
// A layout (doc): lane = M + 16*(K>=2), vgpr = K&1.  C/D: vgpr r -> M=r+8*(lane>=16), N=lane&15.
// ---------------------------------------------------------------------------
__global__ __launch_bounds__(256) void attn_mlp_kernel(
    const float* __restrict__ qbuf, const float* __restrict__ kbuf,
    const float* __restrict__ vbuf, const float* __restrict__ W1,
    const float* __restrict__ b1, const float* __restrict__ W2,
    const float* __restrict__ b2, float* __restrict__ out)
{
    __shared__ float scores[16][64];
    __shared__ float feats[16][32];   // rows 8..15 and cols 24..31 zero pad
    __shared__ float hdn[16][48];
    const int tid = threadIdx.x;

    for (int i = tid; i < 16 * 32; i += 256) (&feats[0][0])[i] = 0.0f;
    __syncthreads();

    const float inv_s12 = 0.28867513459481287f;  // 1/sqrt(12)
    for (int t = tid; t < 1024; t += 256) {
        int m = t >> 6, s = t & 63;              // m = h*8+b
        const float* q = qbuf + m * NQ;
        const float* k = kbuf + (m * 64 + s) * NQ;
        float a = 0.0f;
        #pragma unroll
        for (int d = 0; d < NQ; ++d) a += q[d] * k[d];
        scores[m][s] = a * inv_s12;
    }
    __syncthreads();
    if (tid < 16) {                              // softmax per (h,b) row
        float mx = scores[tid][0];
        for (int s = 1; s < 64; ++s) mx = fmaxf(mx, scores[tid][s]);
        float sum = 0.0f;
        for (int s = 0; s < 64; ++s) { float e = expf(scores[tid][s] - mx); scores[tid][s] = e; sum += e; }
        float inv = 1.0f / sum;
        for (int s = 0; s < 64; ++s) scores[tid][s] *= inv;
    }
    __syncthreads();
    if (tid < 192) {                             // attn @ V
        int m = tid / NQ, d = tid % NQ;
        int h = m >> 3, b = m & 7;
        float a = 0.0f;
        for (int s = 0; s < 64; ++s) a += scores[m][s] * vbuf[(m * 64 + s) * NQ + d];
        feats[b][h * NQ + d] = a;
    }
    __syncthreads();

    const int lane = tid;           // wave 0 only below: EXEC all-ones over 32 lanes
    const int M    = lane & 15;
    const int hiK  = (lane >= 16) ? 2 : 0;
    const int Mr0  = (lane >= 16) ? 8 : 0;

    if (tid < 32) {                 // GEMM1: feats(16x24) @ W1(24x48) + b1, then GELU
        for (int nt = 0; nt < 3; ++nt) {
            const int n0 = nt * 16;
            v8f acc;
            const float cinit = b1[n0 + M];
            #pragma unroll
            for (int r = 0; r < 8; ++r) acc[r] = cinit;
            #pragma unroll
            for (int kk = 0; kk < 6; ++kk) {
                const int k0 = kk * 4 + hiK;
                v2f a, bf;
                a.x  = feats[M][k0];     a.y  = feats[M][k0 + 1];
                bf.x = W1[k0 * 48 + n0 + M];
                bf.y = W1[(k0 + 1) * 48 + n0 + M];
                acc = __builtin_amdgcn_wmma_f32_16x16x4_f32(
                          false, a, false, bf, (short)0, acc, false, false);
            }
            #pragma unroll
            for (int r = 0; r < 8; ++r) {
                float x = acc[r];
                hdn[Mr0 + r][n0 + M] = 0.5f * x * (1.0f + erff(x * 0.70710678118654752f));
            }
        }
    }
    __syncthreads();
    if (tid < 32) {                 // GEMM2: hdn(16x48) @ W2(48x4 padded) + b2
        v8f acc;
        const float c2 = (M < 4) ? b2[M] : 0.0f;
        #pragma unroll
        for (int r = 0; r < 8; ++r) acc[r] = c2;
        #pragma unroll
        for (int kk = 0; kk < 12; ++kk) {
            const int k0 = kk * 4 + hiK;
            v2f a, bf;
            a.x  = hdn[M][k0];  a.y = hdn[M][k0 + 1];
            bf.x = (M < 4) ? W2[k0 * 4 + M] : 0.0f;
            bf.y = (M < 4) ? W2[(k0 + 1) * 4 + M] : 0.0f;
            acc = __builtin_amdgcn_wmma_f32_16x16x4_f32(
                      false, a, false, bf, (short)0, acc, false, false);
        }
        if (lane < 4) {             // D: vgpr r -> row r (lanes 0-15), col = lane
            #pragma unroll
            for (int r = 0; r < 8; ++r) out[r * 4 + lane] = acc[r];
        }
    }
}

// ---------------------------------------------------------------------------
extern "C" void kernel_launch(void* const* d_in, const int* in_sizes, int n_in,
                              void* d_out, int out_size, void* d_ws, size_t ws_size,
                              hipStream_t stream)
{
    const float* seq  = (const float*)d_in[0];   // (8,64)
    const float* resp = (const float*)d_in[1];   // (3,12,3)
    const float* hp   = (const float*)d_in[2];   // (2,3,48)
    const float* W1   = (const float*)d_in[3];   // (24,48)
    const float* b1   = (const float*)d_in[4];   // (48)
    const float* W2   = (const float*)d_in[5];   // (48,4)
    const float* b2   = (const float*)d_in[6];   // (4)

    float* ws     = (float*)d_ws;
    float* states = ws;                          // 512 * 2 * 4096 = 4,194,304 floats
    float* qbuf   = ws + (size_t)512 * 2 * DIM;  // 2*8*12   = 192
    float* kbuf   = qbuf + 192;                  // 2*8*64*12 = 12288
    float* vbuf   = kbuf + 12288;                // 12288

    reservoir_kernel<<<512, 256, 0, stream>>>(seq, resp, states);
    project_kernel<<<2064, 256, 0, stream>>>(states, hp, qbuf, kbuf, vbuf);
    attn_mlp_kernel<<<1, 256, 0, stream>>>(qbuf, kbuf, vbuf, W1, b1, W2, b2,
                                           (float*)d_out);
}